// Transformer_75754633167390
// MI455X (gfx1250) — compile-verified
//
#include <hip/hip_runtime.h>
#include <hip/hip_bf16.h>

typedef __attribute__((ext_vector_type(2))) float v2f;
typedef __attribute__((ext_vector_type(4))) float v4f;
typedef __attribute__((ext_vector_type(8))) float v8f;

// D = A(16x4) * B(4x16) + C, fp32, wave32.
// Fragment layout (ISA 7.12.2):
//   A: lane<16 -> {K=k0,k0+1} of row M=lane; lane>=16 -> {K=k0+2,k0+3} of row M=lane-16
//   B: lane<16 -> {row K=k0,k0+1} at col N=lane; lane>=16 -> {K=k0+2,k0+3} at N=lane-16
//   C/D: VGPR r: lanes0-15 = (M=r, N=lane), lanes16-31 = (M=r+8, N=lane-16)
__device__ __forceinline__ v8f wmma_f32(v2f a, v2f b, v8f c) {
    return __builtin_amdgcn_wmma_f32_16x16x4_f32(false, a, false, b, (short)0, c, false, false);
}

// ---- async global->LDS copies (ASYNCcnt path), guarded: fall back to VGPR staging
#if __has_builtin(__builtin_amdgcn_global_load_async_to_lds_b128) && \
    __has_builtin(__builtin_amdgcn_global_load_async_to_lds_b32) &&  \
    __has_builtin(__builtin_amdgcn_s_wait_asynccnt)
#define HAVE_ASYNC_LDS 1
// Signatures (from compiler diagnostics): (T __device__ *, T __shared__ *, imm int, imm int)
// with T = int4 for b128, T = int for b32.
typedef __attribute__((ext_vector_type(4))) int i4v;
typedef __attribute__((address_space(1))) i4v g_i4;
typedef __attribute__((address_space(3))) i4v l_i4;
typedef __attribute__((address_space(1))) int g_i32;
typedef __attribute__((address_space(3))) int l_i32;
__device__ __forceinline__ void async_cp16(const float* gsrc, float* ldst) {
    __builtin_amdgcn_global_load_async_to_lds_b128((g_i4*)(void*)gsrc, (l_i4*)(void*)ldst, 0, 0);
}
__device__ __forceinline__ void async_cp4(const float* gsrc, float* ldst) {
    __builtin_amdgcn_global_load_async_to_lds_b32((g_i32*)(void*)gsrc, (l_i32*)(void*)ldst, 0, 0);
}
__device__ __forceinline__ void async_wait0() { __builtin_amdgcn_s_wait_asynccnt(0); }
#else
#define HAVE_ASYNC_LDS 0
#endif

// ---------------------------------------------------------------------------
// LDS-tiled GEMM: C[16384,512] = A[16384,512] @ W[512,512] + bias[512]
// Block: 4 waves, 256(M) x 64(N) tile. K in 16-wide LDS chunks, DOUBLE-BUFFERED:
// chunk k+1's global->LDS traffic is issued before chunk k's 64-WMMA compute
// (via async-to-LDS when available), one barrier per chunk.
// Each wave: 4 M-tiles x 4 N-tiles = 16 accumulators.
// head_split=1 writes (B,H,L,64) layout, else row-major.
// ---------------------------------------------------------------------------
#define APITCH 20  // 20*r mod 64 distinct for r=0..15 (no bank conflicts); 80B rows keep 16B align
__global__ __launch_bounds__(128) void gemm512_wmma(const float* __restrict__ A,
                                                    const float* __restrict__ W,
                                                    const float* __restrict__ bias,
                                                    float* __restrict__ C,
                                                    int head_split) {
    __shared__ float As[2][256 * APITCH];  // 2 x 20480 B
    __shared__ float Bs[2][64 * APITCH];   // 2 x  5120 B (transposed W: [n][k])

    const int tid = threadIdx.x;
    const int wave = tid >> 5;
    const int lane = tid & 31;
    const int ml = lane & 15;
    const int hi = lane >> 4;
    const int kh = hi * 2;
    const int m0 = blockIdx.x * 256;
    const int n0 = blockIdx.y * 64;

    // staging geometry
    const int rbase = tid >> 2;       // 0..31   (A rows, 8 reps of 32)
    const int cg = (tid & 3) * 4;     // 0,4,8,12 (A col group)
    const int bk = tid & 15;          // B: k index
    const int bnb = (tid >> 4) * 8;   // B: n group of 8

    const v8f z = {0.f, 0.f, 0.f, 0.f, 0.f, 0.f, 0.f, 0.f};
    v8f acc[4][4];  // [mi][nt]
#pragma unroll
    for (int mi = 0; mi < 4; ++mi)
#pragma unroll
        for (int nt = 0; nt < 4; ++nt) acc[mi][nt] = z;

#if HAVE_ASYNC_LDS
    auto issue_chunk = [&](int buf, int kc) {
#pragma unroll
        for (int rep = 0; rep < 8; ++rep) {
            const int row = rbase + rep * 32;
            async_cp16(A + (size_t)(m0 + row) * 512 + kc + cg, &As[buf][row * APITCH + cg]);
        }
        const float* src = W + (size_t)(kc + bk) * 512 + n0 + bnb;
#pragma unroll
        for (int j = 0; j < 8; ++j) async_cp4(src + j, &Bs[buf][(bnb + j) * APITCH + bk]);
    };
#else
    v4f aReg[8];
    float bReg[8];
    auto load_chunk = [&](int kc) {
#pragma unroll
        for (int rep = 0; rep < 8; ++rep) {
            const int row = rbase + rep * 32;
            aReg[rep] = *(const v4f*)(A + (size_t)(m0 + row) * 512 + kc + cg);
        }
        const float* src = W + (size_t)(kc + bk) * 512 + n0 + bnb;
#pragma unroll
        for (int j = 0; j < 8; ++j) bReg[j] = src[j];
    };
    auto store_chunk = [&](int buf) {
#pragma unroll
        for (int rep = 0; rep < 8; ++rep) {
            const int row = rbase + rep * 32;
            *(v4f*)(&As[buf][row * APITCH + cg]) = aReg[rep];
        }
#pragma unroll
        for (int j = 0; j < 8; ++j) Bs[buf][(bnb + j) * APITCH + bk] = bReg[j];
    };
#endif

    auto compute_chunk = [&](int buf) {
        const float* As_ = As[buf];
        const float* Bs_ = Bs[buf];
#pragma unroll
        for (int kk = 0; kk < 16; kk += 4) {
            v2f bfr[4];
#pragma unroll
            for (int nt = 0; nt < 4; ++nt)
                bfr[nt] = *(const v2f*)(Bs_ + (nt * 16 + ml) * APITCH + kk + kh);
#pragma unroll
            for (int mi = 0; mi < 4; ++mi) {
                const int row = (wave * 4 + mi) * 16 + ml;
                v2f a = *(const v2f*)(As_ + row * APITCH + kk + kh);
#pragma unroll
                for (int nt = 0; nt < 4; ++nt)
                    acc[mi][nt] = wmma_f32(a, bfr[nt], acc[mi][nt]);
            }
        }
    };

    // prologue: fill buffer 0
#if HAVE_ASYNC_LDS
    issue_chunk(0, 0);
    async_wait0();
#else
    load_chunk(0);
    store_chunk(0);
#endif
    __syncthreads();

    for (int ci = 0; ci < 32; ++ci) {
        const int cur = ci & 1;
#if HAVE_ASYNC_LDS
        if (ci + 1 < 32) issue_chunk(cur ^ 1, (ci + 1) * 16);  // overlaps with compute below
        compute_chunk(cur);
        if (ci + 1 < 32) async_wait0();
#else
        if (ci + 1 < 32) load_chunk((ci + 1) * 16);  // loads in flight during compute
        compute_chunk(cur);
        if (ci + 1 < 32) store_chunk(cur ^ 1);       // consumes loads after compute
#endif
        __syncthreads();
    }

    // ---- epilogue
#pragma unroll
    for (int mi = 0; mi < 4; ++mi) {
#pragma unroll
        for (int nt = 0; nt < 4; ++nt) {
            const int n = n0 + nt * 16 + ml;
            const float bv = bias[n];
#pragma unroll
            for (int r = 0; r < 8; ++r) {
                const int m = m0 + (wave * 4 + mi) * 16 + r + hi * 8;
                const float val = acc[mi][nt][r] + bv;
                if (head_split) {
                    // m = b*512 + i ; n = h*64 + c -> [(b*8+h)*512 + i]*64 + c
                    const int bb_ = m >> 9, ii = m & 511, hh = n >> 6, cc = n & 63;
                    C[(((size_t)(bb_ * 8 + hh) * 512) + ii) * 64 + cc] = val;
                } else {
                    C[(size_t)m * 512 + n] = val;
                }
            }
        }
    }
}

// ---------------------------------------------------------------------------
// Keff[b,h,j,d] = sum_{t in window(h), c} Wker[h,d,c,t] * K[b,h,j+t-3,c]
// g[b,h,j]     = sum_{t in window(h), c} bker[h,c,t]  * K[b,h,j+t-3,c]
// One wave per block: 16 rows (j) x 64 cols (d). Skips zero taps per head.
// ---------------------------------------------------------------------------
__global__ void keff_kernel(const float* __restrict__ Kh,
                            const float* __restrict__ Wker,
                            const float* __restrict__ bker,
                            float* __restrict__ Keff,
                            float* __restrict__ g) {
    const int lane = threadIdx.x & 31;
    const int ml = lane & 15;
    const int hi = lane >> 4;
    const int kh = hi * 2;
    const int j0 = blockIdx.x * 16;
    const int bh = blockIdx.y;
    const int h = bh & 7;
    // KWS = {1,1,1,1,3,3,5,7}
    const int kw = (h < 4) ? 1 : (h < 6) ? 3 : (h == 6) ? 5 : 7;
    const int t0 = (7 - kw) >> 1;

    const float* Kbh = Kh + (size_t)bh * 512 * 64;
    const float* Wkh = Wker + (size_t)h * 64 * 64 * 7;  // [d][c][t] strides 448,7,1

    const v8f z = {0.f, 0.f, 0.f, 0.f, 0.f, 0.f, 0.f, 0.f};
    v8f acc[4] = {z, z, z, z};

    for (int t = t0; t < t0 + kw; ++t) {
        const int row = j0 + ml + t - 3;
        const bool ok = ((unsigned)row < 512u);
        const float* arow = Kbh + (size_t)(ok ? row : 0) * 64;
        for (int c0 = 0; c0 < 64; c0 += 4) {
            v2f a = {0.f, 0.f};
            if (ok) a = *(const v2f*)(arow + c0 + kh);
#pragma unroll
            for (int nt = 0; nt < 4; ++nt) {
                const int dd = nt * 16 + ml;
                const float* wp = Wkh + (size_t)dd * 448 + (c0 + kh) * 7 + t;
                v2f b;
                b.x = wp[0];
                b.y = wp[7];
                acc[nt] = wmma_f32(a, b, acc[nt]);
            }
        }
    }

    float* Kebh = Keff + (size_t)bh * 512 * 64;
#pragma unroll
    for (int nt = 0; nt < 4; ++nt) {
#pragma unroll
        for (int r = 0; r < 8; ++r) {
            Kebh[(size_t)(j0 + r + hi * 8) * 64 + nt * 16 + ml] = acc[nt][r];
        }
    }

    // g: lanes 0..15, one query row each
    if (hi == 0) {
        const int j = j0 + ml;
        float s = 0.f;
        for (int t = t0; t < t0 + kw; ++t) {
            const int row = j + t - 3;
            if ((unsigned)row < 512u) {
                const float* kr = Kbh + (size_t)row * 64;
                const float* br = bker + h * 448 + t;  // bker[h,c,t] = [h*448 + c*7 + t]
                for (int c = 0; c < 64; ++c) s += br[c * 7] * kr[c];
            }
        }
        g[(size_t)bh * 512 + j] = s;
    }
}

// ---------------------------------------------------------------------------
// Fused attention per (b,h, 16-row i-tile):
//   S = (Q @ Keff^T + g_j) / 8 ; P = softmax_j(S) ; O[:, h*64: ] = P @ V
// 4 waves / block. Logits 16x512 in LDS (pitch 514). V chunks staged
// transposed in LDS (async when available) so P.V is pure LDS-fed WMMA.
// Per-row biases (Wb,bb,bias_b) dropped: softmax shift-invariant.
// ---------------------------------------------------------------------------
#define LP 514
#define VP 68  // 68*n mod 64 = 4n: distinct banks for n=0..15; even -> b64 aligned
__global__ __launch_bounds__(128) void attn_kernel(const float* __restrict__ Qh,
                                                   const float* __restrict__ Keff,
                                                   const float* __restrict__ gv,
                                                   const float* __restrict__ Vh,
                                                   float* __restrict__ O) {
    __shared__ float Slog[16 * LP];  // 32896 B
    __shared__ float Sq[16 * 66];    //  4224 B
    __shared__ float Vt[64 * VP];    // 17408 B
    __shared__ float red[16 * 8];
    __shared__ float red2[16 * 8];

    const int tid = threadIdx.x;
    const int wave = tid >> 5;
    const int lane = tid & 31;
    const int ml = lane & 15;
    const int hi = lane >> 4;
    const int kh = hi * 2;
    const int i0 = blockIdx.x * 16;
    const int bh = blockIdx.y;
    const int b = bh >> 3, h = bh & 7;

    const float* Qbh = Qh + (size_t)bh * 512 * 64;
    const float* Kebh = Keff + (size_t)bh * 512 * 64;
    const float* Vbh = Vh + (size_t)bh * 512 * 64;
    const float* gbh = gv + (size_t)bh * 512;

    // stage Q tile (16x64)
    for (int idx = tid; idx < 16 * 64; idx += 128) {
        const int r = idx >> 6, c = idx & 63;
        Sq[r * 66 + c] = Qbh[(size_t)(i0 + r) * 64 + c];
    }
    __syncthreads();

    // ---- S = Q @ Keff^T : each wave handles 8 j-tiles ----
    for (int jt = wave * 8; jt < wave * 8 + 8; ++jt) {
        const int j0 = jt * 16;
        v8f acc = {0.f, 0.f, 0.f, 0.f, 0.f, 0.f, 0.f, 0.f};
        const float* krow = Kebh + (size_t)(j0 + ml) * 64;
#pragma unroll
        for (int kk = 0; kk < 64; kk += 4) {
            v2f a = *(const v2f*)(Sq + ml * 66 + kk + kh);
            v2f bfr = *(const v2f*)(krow + kk + kh);
            acc = wmma_f32(a, bfr, acc);
        }
        const float gj = gbh[j0 + ml];
#pragma unroll
        for (int r = 0; r < 8; ++r) {
            Slog[(r + hi * 8) * LP + j0 + ml] = (acc[r] + gj) * 0.125f;
        }
    }
    __syncthreads();

    // ---- softmax over j (512): 8 threads per row, 64 cols each ----
    {
        const int r = tid >> 3, seg = tid & 7;
        float* row = Slog + r * LP + seg * 64;
        float mx = -3.4e38f;
        for (int j = 0; j < 64; ++j) mx = fmaxf(mx, row[j]);
        red[r * 8 + seg] = mx;
        __syncthreads();
        mx = red[r * 8];
        for (int s = 1; s < 8; ++s) mx = fmaxf(mx, red[r * 8 + s]);
        float sum = 0.f;
        for (int j = 0; j < 64; ++j) {
            const float e = __expf(row[j] - mx);
            row[j] = e;
            sum += e;
        }
        red2[r * 8 + seg] = sum;
        __syncthreads();
        float tot = 0.f;
        for (int s = 0; s < 8; ++s) tot += red2[r * 8 + s];
        const float inv = 1.0f / tot;
        for (int j = 0; j < 64; ++j) row[j] *= inv;
    }
    __syncthreads();

    // ---- O = P @ V : V chunks staged transposed; each wave owns one N tile ----
    {
        const int n = wave * 16 + ml;
        const int vk = tid >> 1;        // 0..63
        const int vcb = (tid & 1) * 32; // 0 or 32
        v8f acc = {0.f, 0.f, 0.f, 0.f, 0.f, 0.f, 0.f, 0.f};
        for (int kc = 0; kc < 512; kc += 64) {
            // stage V[kc..kc+63][0..63] -> Vt[n][k]
            const float* src = Vbh + (size_t)(kc + vk) * 64 + vcb;
#if HAVE_ASYNC_LDS
#pragma unroll
            for (int j = 0; j < 32; ++j) async_cp4(src + j, &Vt[(vcb + j) * VP + vk]);
            async_wait0();
#else
#pragma unroll
            for (int j = 0; j < 32; ++j) Vt[(vcb + j) * VP + vk] = src[j];
#endif
            __syncthreads();
#pragma unroll
            for (int kk = 0; kk < 64; kk += 4) {
                v2f a = *(const v2f*)(Slog + ml * LP + kc + kk + kh);
                v2f bfr = *(const v2f*)(Vt + n * VP + kk + kh);
                acc = wmma_f32(a, bfr, acc);
            }
            __syncthreads();
        }
#pragma unroll
        for (int r = 0; r < 8; ++r) {
            const int row = i0 + r + hi * 8;
            O[((size_t)(b * 512 + row)) * 512 + h * 64 + n] = acc[r];
        }
    }
}

// ---------------------------------------------------------------------------
extern "C" void kernel_launch(void* const* d_in, const int* in_sizes, int n_in,
                              void* d_out, int out_size, void* d_ws, size_t ws_size,
                              hipStream_t stream) {
    (void)in_sizes; (void)n_in; (void)out_size; (void)ws_size;
    const float* q    = (const float*)d_in[0];
    const float* k    = (const float*)d_in[1];
    const float* v    = (const float*)d_in[2];
    const float* Wq   = (const float*)d_in[3];
    const float* bq   = (const float*)d_in[4];
    const float* Wk   = (const float*)d_in[5];
    const float* bk   = (const float*)d_in[6];
    const float* Wv   = (const float*)d_in[7];
    const float* bv   = (const float*)d_in[8];
    const float* Wker = (const float*)d_in[9];
    const float* bker = (const float*)d_in[10];
    // d_in[11]=Wb, d_in[12]=bb, d_in[13]=bias_b: softmax-shift-invariant, unused.
    const float* Wp   = (const float*)d_in[14];
    const float* bp   = (const float*)d_in[15];
    float* out = (float*)d_out;
    float* ws = (float*)d_ws;

    const size_t HEADS = 8388608;  // 32*8*512*64 floats
    float* Qh   = ws;
    float* Kh   = ws + HEADS;
    float* Vh   = ws + 2 * HEADS;
    float* Keff = ws + 3 * HEADS;
    float* g    = ws + 4 * HEADS;
    float* O    = Kh;  // Kh dead after keff_kernel; reuse for attention output

    const dim3 gg(64, 8), gb(128);
    gemm512_wmma<<<gg, gb, 0, stream>>>(q, Wq, bq, Qh, 1);
    gemm512_wmma<<<gg, gb, 0, stream>>>(k, Wk, bk, Kh, 1);
    gemm512_wmma<<<gg, gb, 0, stream>>>(v, Wv, bv, Vh, 1);
    keff_kernel<<<dim3(32, 256), 32, 0, stream>>>(Kh, Wker, bker, Keff, g);
    attn_kernel<<<dim3(32, 256), 128, 0, stream>>>(Qh, Keff, g, Vh, O);
    gemm512_wmma<<<gg, gb, 0, stream>>>(O, Wp, bp, out, 0);
}